// GCN_20830591385878
// MI455X (gfx1250) — compile-verified
//
#include <hip/hip_runtime.h>
#include <hip/hip_bf16.h>
#include <stdint.h>

typedef __attribute__((ext_vector_type(2))) float v2f;
typedef __attribute__((ext_vector_type(8))) float v8f;
typedef __attribute__((ext_vector_type(4))) unsigned int v4u;
typedef __attribute__((ext_vector_type(4))) int v4i;
typedef __attribute__((ext_vector_type(8))) int v8i;

#define NROWS 2048
#define FDIM  256
#define ODIM  256
#define BATCH 8
#define MTOT  (BATCH * NROWS)   // 16384

// ---- TDM availability probe (device pass only; host pass takes fallback) ----
#if defined(__HIP_DEVICE_COMPILE__) && defined(__has_builtin)
#if __has_builtin(__builtin_amdgcn_tensor_load_to_lds) && __has_builtin(__builtin_amdgcn_s_wait_tensorcnt)
#define USE_TDM 1
#endif
#endif
#ifndef USE_TDM
#define USE_TDM 0
#endif

#if USE_TDM
// D# group1 words for the X tile: 128 rows x 32 DWORDs, tensor row stride 256,
// data_size=4B (code 2), pad_enable=1, pad_interval=32 DW (code 4),
// pad_amount=4 DW (code 3)  ->  LDS row pitch = 36 floats (bank-conflict-free).
#define XG1_W0 ((2 << 16) | (1 << 20) | (4 << 22) | (3 << 25))
#define XG1_W1 (32 << 16)     // tensor_dim0 = 32
#define XG1_W2 (128 << 16)    // tensor_dim1 = 128
#define XG1_W3 (32 << 16)     // tile_dim0 = 32
#define XG1_W4 (128)          // tile_dim1 = 128
#define XG1_W5 (256)          // tensor_dim0_stride = 256

// D# group1 words for the W chunk: contiguous 4096 DWORDs (1D), no padding.
#define WG1_W0 (2 << 16)      // data_size = 4B
#define WG1_W1 (4096 << 16)   // tensor_dim0 = 4096
#define WG1_W2 (0)
#define WG1_W3 (4096 << 16)   // tile_dim0 = 4096
#define WG1_W4 (0)            // tile_dim1 = 0 (unused -> 1D)
#define WG1_W5 (4096)         // tensor_dim0_stride (unused for 1D)

__device__ __forceinline__ void tdm_issue(unsigned lds_off, unsigned long long gaddr,
                                          int w0, int w1, int w2, int w3, int w4, int w5) {
    v4u g0;
    g0.x = 1u;                                               // count=1, user mode
    g0.y = lds_off;                                          // lds_addr
    g0.z = (unsigned)gaddr;                                  // global_addr[31:0]
    g0.w = (unsigned)((gaddr >> 32) & 0x1FFFFFFull) | (2u << 30); // [56:32] + type=2
    v8i g1 = { w0, w1, w2, w3, w4, w5, 0, 0 };
    v4i z4 = {};
#if __clang_major__ >= 23
    v8i z8 = {};
    __builtin_amdgcn_tensor_load_to_lds(g0, g1, z4, z4, z8, 0);
#else
    __builtin_amdgcn_tensor_load_to_lds(g0, g1, z4, z4, 0);
#endif
}
#endif  // USE_TDM

// ---------------------------------------------------------------------------
// Kernel 1: s1[row] = x[row,:]·w1, s2[row] = x[row,:]·w2  (one wave per row)
// ---------------------------------------------------------------------------
__global__ __launch_bounds__(256) void gcn_s12(const float* __restrict__ x,
                                               const float* __restrict__ adj_w,
                                               float* __restrict__ s1,
                                               float* __restrict__ s2) {
    const int w = threadIdx.x >> 5, lane = threadIdx.x & 31;
    const int row = blockIdx.x * 8 + w;
    const float4* xr = (const float4*)(x + (size_t)row * FDIM);
    const float4* w1 = (const float4*)(adj_w);
    const float4* w2 = (const float4*)(adj_w + FDIM);
    float a1 = 0.f, a2 = 0.f;
#pragma unroll
    for (int i = 0; i < 2; ++i) {
        int idx = i * 32 + lane;
        float4 xv = xr[idx];
        float4 u = w1[idx], t = w2[idx];
        a1 += xv.x * u.x + xv.y * u.y + xv.z * u.z + xv.w * u.w;
        a2 += xv.x * t.x + xv.y * t.y + xv.z * t.z + xv.w * t.w;
    }
#pragma unroll
    for (int off = 16; off > 0; off >>= 1) {
        a1 += __shfl_xor(a1, off, 32);
        a2 += __shfl_xor(a2, off, 32);
    }
    if (lane == 0) { s1[row] = a1; s2[row] = a2; }
}

// ---------------------------------------------------------------------------
// Kernel 2: per-batch sum of s2 (deterministic block reduction)
// ---------------------------------------------------------------------------
__global__ __launch_bounds__(256) void gcn_s2sum(const float* __restrict__ s2,
                                                 float* __restrict__ s2sum) {
    __shared__ float red[256];
    const int b = blockIdx.x, t = threadIdx.x;
    float a = 0.f;
    for (int j = t; j < NROWS; j += 256) a += s2[b * NROWS + j];
    red[t] = a; __syncthreads();
    for (int s = 128; s > 0; s >>= 1) {
        if (t < s) red[t] += red[t + s];
        __syncthreads();
    }
    if (t == 0) s2sum[b] = red[0];
}

// ---------------------------------------------------------------------------
// Kernel 3: dinv[i] = rsqrt(max(N*(s1+c) + sum(s2) + 1, 1))
// ---------------------------------------------------------------------------
__global__ __launch_bounds__(256) void gcn_dinv(const float* __restrict__ s1,
                                                const float* __restrict__ s2sum,
                                                const float* __restrict__ adj_b,
                                                float* __restrict__ dinv) {
    const int i = blockIdx.x * 256 + threadIdx.x;
    const float c = adj_b[0];
    float deg = (float)NROWS * (s1[i] + c) + s2sum[i >> 11] + 1.0f;
    deg = fmaxf(deg, 1.0f);
    dinv[i] = rsqrtf(deg);
}

// ---------------------------------------------------------------------------
// Kernel 4: permute W into WMMA B-fragment lane layout.
// pair p = kt*4096 + (nt*8 + ksl)*32 + lane ;
// lanes 0-15 -> {W[k][n], W[k+1][n]}, lanes 16-31 -> K+2.
// ---------------------------------------------------------------------------
__global__ __launch_bounds__(256) void gcn_wperm(const float* __restrict__ W,
                                                 float* __restrict__ Wp) {
    const int p = blockIdx.x * 256 + threadIdx.x;   // 0..32767
    const int lane = p & 31;
    const int q = p >> 5;
    const int nksl = q & 127;
    const int kt = q >> 7;
    const int nt = nksl >> 3, ksl = nksl & 7;
    const int k = (kt * 8 + ksl) * 4 + ((lane >> 4) << 1);
    const int n = nt * 16 + (lane & 15);
    float2 val;
    val.x = W[(size_t)k * ODIM + n];
    val.y = W[(size_t)(k + 1) * ODIM + n];
    ((float2*)Wp)[p] = val;
}

// ---------------------------------------------------------------------------
// Kernel 5: H = x @ W via V_WMMA_F32_16X16X4_F32, TDM double-buffered staging.
// Block: 256 threads (8 waves), tile 128 rows x 128 cols; grid (128, 2).
// ---------------------------------------------------------------------------
__global__ __launch_bounds__(256) void gcn_gemm(const float* __restrict__ x,
                                                const float* __restrict__ Wp,
                                                float* __restrict__ H) {
#if USE_TDM
    __shared__ float Xs[2][128 * 36];
    __shared__ float Ws[2][4096];
#else
    __shared__ float Xs[1][128 * 36];
    __shared__ float Ws[1][4096];
#endif
    const int tid = threadIdx.x;
    const int w = tid >> 5, lane = tid & 31;
    const int lr = lane & 15;
    const int hi = lane >> 4;
    const int m0 = blockIdx.x * 128;
    const int ny = blockIdx.y;

    v8f acc[8];
#pragma unroll
    for (int i = 0; i < 8; ++i) acc[i] = (v8f){};

#if USE_TDM
    const unsigned xs_off0 = (unsigned)(uintptr_t)&Xs[0][0];
    const unsigned xs_off1 = (unsigned)(uintptr_t)&Xs[1][0];
    const unsigned ws_off0 = (unsigned)(uintptr_t)&Ws[0][0];
    const unsigned ws_off1 = (unsigned)(uintptr_t)&Ws[1][0];
    const unsigned long long xbase = (unsigned long long)(uintptr_t)(x + (size_t)m0 * FDIM);
    const unsigned long long wbase = (unsigned long long)(uintptr_t)(Wp + ny * 4096);
    if (tid < 32) {   // wave 0 only (uniform branch; TDM ignores EXEC)
        tdm_issue(xs_off0, xbase, XG1_W0, XG1_W1, XG1_W2, XG1_W3, XG1_W4, XG1_W5);
        tdm_issue(ws_off0, wbase, WG1_W0, WG1_W1, WG1_W2, WG1_W3, WG1_W4, WG1_W5);
    }
#endif

    for (int kt = 0; kt < 8; ++kt) {
#if USE_TDM
        const int cur = kt & 1;
        if (tid < 32) {
            if (kt < 7) {
                // prefetch next chunk into the other buffer, then wait for the
                // two older (current-buffer) TDM ops: in-order => tensorcnt<=2.
                const unsigned xo = cur ? xs_off0 : xs_off1;
                const unsigned wo = cur ? ws_off0 : ws_off1;
                tdm_issue(xo, xbase + (unsigned long long)(kt + 1) * 128u,
                          XG1_W0, XG1_W1, XG1_W2, XG1_W3, XG1_W4, XG1_W5);
                tdm_issue(wo, wbase + (unsigned long long)(kt + 1) * 32768u,
                          WG1_W0, WG1_W1, WG1_W2, WG1_W3, WG1_W4, WG1_W5);
                __builtin_amdgcn_s_wait_tensorcnt(2);
            } else {
                __builtin_amdgcn_s_wait_tensorcnt(0);
            }
        }
        __syncthreads();
        const float* Xc = Xs[cur];
        const float* Wc = Ws[cur];
#else
        // Manual staging fallback (coalesced, bank-conflict-free LDS layout).
        {
            const int rr = tid >> 4;
            const int k2 = tid & 15;
#pragma unroll
            for (int pass = 0; pass < 8; ++pass) {
                const int r = pass * 16 + rr;
                float2 xv = *(const float2*)(x + (size_t)(m0 + r) * FDIM + kt * 32 + k2 * 2);
                *(float2*)&Xs[0][r * 36 + k2 * 2] = xv;
            }
            const float4* src = (const float4*)(Wp + kt * 8192 + ny * 4096);
#pragma unroll
            for (int i = 0; i < 4; ++i) {
                const int idx = i * 256 + tid;
                ((float4*)&Ws[0][0])[idx] = src[idx];
            }
        }
        __syncthreads();
        const float* Xc = Xs[0];
        const float* Wc = Ws[0];
#endif

#pragma unroll
        for (int ksl = 0; ksl < 8; ++ksl) {
            v2f a = *(const v2f*)&Xc[(w * 16 + lr) * 36 + ksl * 4 + hi * 2];
#pragma unroll
            for (int nt = 0; nt < 8; ++nt) {
                v2f b = *(const v2f*)&Wc[(nt * 8 + ksl) * 64 + lane * 2];
                acc[nt] = __builtin_amdgcn_wmma_f32_16x16x4_f32(
                    false, a, false, b, (short)0, acc[nt], false, false);
            }
        }
        __syncthreads();
    }

    // Epilogue: C/D layout -> vgpr v holds row v + 8*hi, col lr of each n-tile.
#pragma unroll
    for (int nt = 0; nt < 8; ++nt) {
        const int col = (ny * 8 + nt) * 16 + lr;
#pragma unroll
        for (int v = 0; v < 8; ++v) {
            const int row = m0 + w * 16 + v + hi * 8;
            H[(size_t)row * ODIM + col] = acc[nt][v];
        }
    }
}

// ---------------------------------------------------------------------------
// Kernel 6: partial u/v reductions (deterministic, no atomics)
// ---------------------------------------------------------------------------
__global__ __launch_bounds__(256) void gcn_uv_part(const float* __restrict__ H,
                                                   const float* __restrict__ dinv,
                                                   const float* __restrict__ s2,
                                                   float* __restrict__ pu,
                                                   float* __restrict__ pv) {
    __shared__ float dl[256], el[256];
    const int b = blockIdx.x >> 3;
    const int jt = blockIdx.x & 7;
    const int t = threadIdx.x;
    const int rowbase = b * NROWS + jt * 256;
    const float d = dinv[rowbase + t];
    dl[t] = d;
    el[t] = d * s2[rowbase + t];
    __syncthreads();
    float up = 0.f, vp = 0.f;
    for (int j = 0; j < 256; ++j) {
        float h = H[(size_t)(rowbase + j) * ODIM + t];
        up += dl[j] * h;
        vp += el[j] * h;
    }
    pu[(size_t)blockIdx.x * ODIM + t] = up;
    pv[(size_t)blockIdx.x * ODIM + t] = vp;
}

__global__ __launch_bounds__(256) void gcn_uv_final(const float* __restrict__ pu,
                                                    const float* __restrict__ pv,
                                                    float* __restrict__ u,
                                                    float* __restrict__ v) {
    const int b = blockIdx.x, t = threadIdx.x;
    float su = 0.f, sv = 0.f;
#pragma unroll
    for (int jt = 0; jt < 8; ++jt) {
        su += pu[(b * 8 + jt) * ODIM + t];
        sv += pv[(b * 8 + jt) * ODIM + t];
    }
    u[b * ODIM + t] = su;
    v[b * ODIM + t] = sv;
}

// ---------------------------------------------------------------------------
// Kernel 7: out[i,o] = relu(dinv*((s1+c)*u + v) + dinv^2*H + bias), in place.
// ---------------------------------------------------------------------------
__global__ __launch_bounds__(256) void gcn_final(float* __restrict__ H,   // = d_out
                                                 const float* __restrict__ s1,
                                                 const float* __restrict__ dinv,
                                                 const float* __restrict__ adj_b,
                                                 const float* __restrict__ u,
                                                 const float* __restrict__ v,
                                                 const float* __restrict__ bias) {
    const int row = blockIdx.x;
    const int o = threadIdx.x;
    const int b = row >> 11;
    const float c = adj_b[0];
    const float di = dinv[row];
    const float a = di * (s1[row] + c);
    const size_t idx = (size_t)row * ODIM + o;
    const float h = H[idx];
    float r = a * u[b * ODIM + o] + di * v[b * ODIM + o] + di * di * h + bias[o];
    H[idx] = fmaxf(r, 0.f);
}

// ---------------------------------------------------------------------------
extern "C" void kernel_launch(void* const* d_in, const int* in_sizes, int n_in,
                              void* d_out, int out_size, void* d_ws, size_t ws_size,
                              hipStream_t stream) {
    (void)in_sizes; (void)n_in; (void)out_size; (void)ws_size;
    const float* x     = (const float*)d_in[0];
    const float* adj_w = (const float*)d_in[1];
    const float* adj_b = (const float*)d_in[2];
    const float* W     = (const float*)d_in[3];
    const float* bias  = (const float*)d_in[4];
    float* out = (float*)d_out;
    float* ws  = (float*)d_ws;

    float* s1    = ws;            // 16384
    float* s2    = ws + 16384;    // 16384
    float* dinv  = ws + 32768;    // 16384
    float* s2sum = ws + 49152;    // 8 (+pad)
    float* u     = ws + 49216;    // 2048
    float* v     = ws + 51264;    // 2048
    float* pu    = ws + 53312;    // 16384
    float* pv    = ws + 69696;    // 16384
    float* Wp    = ws + 86080;    // 65536  (total ~593 KB)

    gcn_wperm<<<128, 256, 0, stream>>>(W, Wp);
    gcn_s12<<<2048, 256, 0, stream>>>(x, adj_w, s1, s2);
    gcn_s2sum<<<8, 256, 0, stream>>>(s2, s2sum);
    gcn_dinv<<<64, 256, 0, stream>>>(s1, s2sum, adj_b, dinv);
    gcn_gemm<<<dim3(128, 2), 256, 0, stream>>>(x, Wp, out);
    gcn_uv_part<<<64, 256, 0, stream>>>(out, dinv, s2, pu, pv);
    gcn_uv_final<<<8, 256, 0, stream>>>(pu, pv, u, v);
    gcn_final<<<16384, 256, 0, stream>>>(out, s1, dinv, adj_b, u, v, bias);
}